// PatchLevelPruner_54271206752568
// MI455X (gfx1250) — compile-verified
//
#include <hip/hip_runtime.h>
#include <hip/hip_bf16.h>
#include <math.h>

typedef float v2f __attribute__((ext_vector_type(2)));
typedef float v8f __attribute__((ext_vector_type(8)));
typedef int   v4i __attribute__((vector_size(16)));   // type expected by async builtin
typedef __attribute__((address_space(1))) v4i gv4i;   // global int4
typedef __attribute__((address_space(3))) v4i lv4i;   // LDS int4

#define C_IN    256
#define C_HID   64
#define BSZ     8
#define WIMG    128
#define NTOK    (128 * 128)          // tokens per batch
#define NP      1024                 // patches per batch (32x32)
#define KKEEP   716                  // int(1024*0.7), clamped [10,1024]
#define TPP     16                   // tokens per patch (4x4)
#define AROW2   132                  // padded LDS row stride for half-tiles (128+4)
#define BUFF    (TPP * AROW2)        // 2112 floats per half-tile buffer
#define WAVES   4
#define SCORE_OFF ((size_t)BSZ * KKEEP * TPP * C_IN)

#if defined(__AMDGCN__) && __has_builtin(__builtin_amdgcn_global_load_async_to_lds_b128)
#define HAS_ASYNC 1
#else
#define HAS_ASYNC 0
#endif

__device__ __forceinline__ void wait_async_16() {
#if __has_builtin(__builtin_amdgcn_s_wait_asynccnt)
    __builtin_amdgcn_s_wait_asynccnt(16);
#else
    asm volatile("s_wait_asynccnt 0x10" ::: "memory");
#endif
}
__device__ __forceinline__ void wait_async_0() {
#if __has_builtin(__builtin_amdgcn_s_wait_asynccnt)
    __builtin_amdgcn_s_wait_asynccnt(0);
#else
    asm volatile("s_wait_asynccnt 0x0" ::: "memory");
#endif
}

// Resolve patch coordinates for (global wave, iteration)
__device__ __forceinline__ const float* patch_base(const float* tokens, int gw, int it,
                                                   int* pi, int* pj) {
    int P = gw * 4 + it;
    int b = P >> 10, p = P & (NP - 1);
    *pi = p >> 5; *pj = p & 31;
    return tokens + (size_t)b * NTOK * C_IN;
}

// Stage one K-half (16 tokens x 128 f32) of a patch tile into an LDS buffer.
// 512 float4s -> 16 async b128 ops/lane; each op moves one full padded row
// slice (coalesced 512B per instruction), tracked by ASYNCcnt.
__device__ __forceinline__ void stage_async(const float* __restrict__ tb, float* buf,
                                            int lane, int pi, int pj, int h) {
#if HAS_ASYNC
#pragma unroll
    for (int i = 0; i < 16; ++i) {
        int idx = i * 32 + lane;
        int row = idx >> 5, col4 = idx & 31;          // 32 float4 per row-half
        int n = (pi * 4 + (row >> 2)) * WIMG + pj * 4 + (row & 3);
        const float* gp = tb + (size_t)n * C_IN + h * 128 + col4 * 4;
        float* lp = buf + row * AROW2 + col4 * 4;
        __builtin_amdgcn_global_load_async_to_lds_b128((gv4i*)gp, (lv4i*)lp, 0, 0);
    }
#endif
}

__device__ __forceinline__ void stage_sync(const float* __restrict__ tb, float* buf,
                                           int lane, int pi, int pj, int h) {
#pragma unroll
    for (int i = 0; i < 16; ++i) {
        int idx = i * 32 + lane;
        int row = idx >> 5, col4 = idx & 31;
        int n = (pi * 4 + (row >> 2)) * WIMG + pj * 4 + (row & 3);
        float4 v = *(const float4*)(tb + (size_t)n * C_IN + h * 128 + col4 * 4);
        *(float4*)(buf + row * AROW2 + col4 * 4) = v;
    }
}

// ---------------------------------------------------------------------------
// Kernel 1: per-patch importance scores via f32 WMMA with an async-LDS
// double-buffered pipeline (2 half-tiles in flight per wave).
// ---------------------------------------------------------------------------
__global__ __launch_bounds__(32 * WAVES)
void k_scores(const float* __restrict__ tokens,
              const float* __restrict__ w1, const float* __restrict__ b1,
              const float* __restrict__ w2, const float* __restrict__ b2,
              float* __restrict__ scores)
{
    __shared__ float sB[C_IN * C_HID];          // 64 KB: w1 in WMMA-B-fragment order
    __shared__ float sA[WAVES * 2 * BUFF];      // 66 KB: per-wave double buffers
    __shared__ float sb1[C_HID];
    __shared__ float sw2[C_HID];
    __shared__ float sb2s[1];

    const int tid = threadIdx.x;

    // Stage w1 into WMMA-B-ready layout: chunk (kg,nt) = 64 floats, lane l's
    // float2 at [2l]: [2l+half] = w1[4kg + 2*(l/16) + half][16nt + (l&15)]
    for (int e = tid; e < C_IN * C_HID; e += 32 * WAVES) {
        int chunk = e >> 6;
        int i     = e & 63;
        int l     = i >> 1;
        int half  = i & 1;
        int kg    = chunk >> 2;
        int nt    = chunk & 3;
        int krow  = kg * 4 + ((l >> 4) << 1) + half;
        int ncol  = nt * 16 + (l & 15);
        sB[e] = w1[krow * C_HID + ncol];
    }
    if (tid < C_HID) { sb1[tid] = b1[tid]; sw2[tid] = w2[tid]; }
    if (tid == 0)    { sb2s[0] = b2[0]; }
    __syncthreads();

    const int lane = tid & 31;
    const int wav  = tid >> 5;
    const int m    = lane & 15;          // A fragment: M = lane%16
    const int kh   = (lane >> 4) << 1;   // lanes 16-31 hold K+2,K+3
    const float b2v = sb2s[0];
    float* buf0 = &sA[(wav * 2 + 0) * BUFF];
    float* buf1 = &sA[(wav * 2 + 1) * BUFF];

    const int gw = blockIdx.x * WAVES + wav;   // 2048 waves, 4 patches each

#if HAS_ASYNC
    {   // prologue: stages 0 and 1 (patch 0, halves 0/1)
        int pi0, pj0;
        const float* tb0 = patch_base(tokens, gw, 0, &pi0, &pj0);
        stage_async(tb0, buf0, lane, pi0, pj0, 0);
        stage_async(tb0, buf1, lane, pi0, pj0, 1);
    }
#endif

    for (int it = 0; it < 4; ++it) {
        const int P = gw * 4 + it;
        int pi, pj;
        const float* tb = patch_base(tokens, gw, it, &pi, &pj);
        (void)tb;

        v8f acc[4] = {};
#pragma unroll
        for (int h = 0; h < 2; ++h) {
            const int s = it * 2 + h;           // pipeline stage 0..7
            float* A = (s & 1) ? buf1 : buf0;
#if HAS_ASYNC
            if (s < 7) wait_async_16();         // oldest stage complete (in-order)
            else       wait_async_0();
#else
            stage_sync(tb, A, lane, pi, pj, h); // LDS ops in-order within wave
#endif
            for (int j = 0; j < 32; ++j) {
                const int kg = h * 32 + j;
                v2f a = *(const v2f*)(A + m * AROW2 + j * 4 + kh);
                const float* bp = &sB[(kg * 4) * 64 + lane * 2];
#pragma unroll
                for (int nt = 0; nt < 4; ++nt) {
                    v2f bb = *(const v2f*)(bp + nt * 64);
                    acc[nt] = __builtin_amdgcn_wmma_f32_16x16x4_f32(
                        false, a, false, bb, (short)0, acc[nt], false, false);
                }
            }
#if HAS_ASYNC
            if (s + 2 < 8) {                    // refill the buffer just consumed
                const int s2 = s + 2, it2 = s2 >> 1, h2 = s2 & 1;
                int pi2, pj2;
                const float* tb2 = patch_base(tokens, gw, it2, &pi2, &pj2);
                stage_async(tb2, A, lane, pi2, pj2, h2);
            }
#endif
        }

        // Layer 2: s[g] = sum_n gelu(h[M][n]+b1[n]) * w2[n]; lane owns one column.
        float s[8];
#pragma unroll
        for (int g = 0; g < 8; ++g) s[g] = 0.0f;
#pragma unroll
        for (int nt = 0; nt < 4; ++nt) {
            float bb1 = sb1[nt * 16 + m];
            float ww2 = sw2[nt * 16 + m];
#pragma unroll
            for (int g = 0; g < 8; ++g) {
                float x  = acc[nt][g] + bb1;
                float ge = 0.5f * x * (1.0f + erff(x * 0.70710678118654752f));
                s[g] += ge * ww2;
            }
        }
#pragma unroll
        for (int g = 0; g < 8; ++g) {           // reduce within 16-lane halves
            s[g] += __shfl_xor(s[g], 1, 32);
            s[g] += __shfl_xor(s[g], 2, 32);
            s[g] += __shfl_xor(s[g], 4, 32);
            s[g] += __shfl_xor(s[g], 8, 32);
        }
        float ps = 0.0f;
#pragma unroll
        for (int g = 0; g < 8; ++g) {
            float t = s[g] + b2v;
            ps += 1.0f / (1.0f + expf(-t));      // sigmoid per token
        }
        ps += __shfl_xor(ps, 16, 32);            // combine the two 8-token halves
        if (lane == 0) scores[P] = ps * 0.0625f; // mean over 16 tokens
    }
}

// ---------------------------------------------------------------------------
// Kernel 2: per-batch top-K (jax.lax.top_k tie semantics), compacted ascending.
// ---------------------------------------------------------------------------
__global__ __launch_bounds__(NP)
void k_select(const float* __restrict__ scores, int* __restrict__ keep)
{
    __shared__ float s[NP];
    __shared__ int wtot[NP / 32];
    const int b = blockIdx.x;
    const int p = threadIdx.x;
    s[p] = scores[b * NP + p];
    __syncthreads();

    const float my = s[p];
    int rank = 0;
    for (int q = 0; q < NP; ++q) {
        float v = s[q];
        rank += (v > my) || ((v == my) && (q < p));
    }
    const bool kept = rank < KKEEP;   // ranks distinct -> exactly KKEEP kept

    unsigned long long mball = __ballot(kept);
    const int lane = p & 31, wid = p >> 5;
    int lpre = __popcll(mball & ((1ull << lane) - 1ull));
    if (lane == 0) wtot[wid] = __popcll(mball);
    __syncthreads();
    int pre = 0;
    for (int w0 = 0; w0 < wid; ++w0) pre += wtot[w0];
    if (kept) keep[b * NP + pre + lpre] = p;
}

// ---------------------------------------------------------------------------
// Kernel 3: gather kept patches -> pruned_tokens [B, K*16, C], float4 copies.
// ---------------------------------------------------------------------------
__global__ __launch_bounds__(256)
void k_gather(const float* __restrict__ tokens, const int* __restrict__ keep,
              float* __restrict__ out)
{
    const int bk = blockIdx.x;           // 0 .. B*K-1
    const int b  = bk / KKEEP;
    const int k  = bk - b * KKEEP;
    const int p  = keep[b * NP + k];
    const int pi = p >> 5, pj = p & 31;

    const float4* src = (const float4*)tokens + (size_t)b * NTOK * (C_IN / 4);
    float4*       dst = (float4*)out + (size_t)bk * TPP * (C_IN / 4);

    for (int e = threadIdx.x; e < TPP * (C_IN / 4); e += 256) {
        int t = e >> 6, col = e & 63;
        int n = (pi * 4 + (t >> 2)) * WIMG + pj * 4 + (t & 3);
        dst[(size_t)t * (C_IN / 4) + col] = src[(size_t)n * (C_IN / 4) + col];
    }
}

// ---------------------------------------------------------------------------
extern "C" void kernel_launch(void* const* d_in, const int* in_sizes, int n_in,
                              void* d_out, int out_size, void* d_ws, size_t ws_size,
                              hipStream_t stream)
{
    const float* tokens = (const float*)d_in[0];
    const float* w1     = (const float*)d_in[1];
    const float* b1     = (const float*)d_in[2];
    const float* w2     = (const float*)d_in[3];
    const float* b2     = (const float*)d_in[4];

    float* out    = (float*)d_out;
    float* scores = out + SCORE_OFF;     // patch_scores section of d_out
    int*   keep   = (int*)d_ws;          // B*NP ints of scratch

    k_scores<<<(BSZ * NP) / (WAVES * 4), 32 * WAVES, 0, stream>>>(
        tokens, w1, b1, w2, b2, scores);
    k_select<<<BSZ, NP, 0, stream>>>(scores, keep);
    k_gather<<<BSZ * KKEEP, 256, 0, stream>>>(tokens, keep, out);
}